// ContrastiveLoss_70257075028025
// MI455X (gfx1250) — compile-verified
//
#include <hip/hip_runtime.h>
#include <hip/hip_bf16.h>
#include <math.h>

typedef __attribute__((ext_vector_type(2))) float v2f;
typedef __attribute__((ext_vector_type(8))) float v8f;

#define BB   4
#define CC   64
#define HWP  4096      // 64*64
#define NEGK 256
#define TEMPR 2.0f
#define EPSV 1e-8f
#define INV_MAX_EUC 0.011223988f   // 1/sqrt(63^2+63^2)
#define INV_MAX_RGB 0.57735026919f // 1/sqrt(3)

// ---------------------------------------------------------------- zero accums
__global__ void k_zero(float* p, int n) {
    int i = blockIdx.x * 256 + threadIdx.x;
    if (i < n) p[i] = 0.0f;
}

// ------------------------------------------------- (B,C,HW) -> (B,HW,C) f32
__global__ void k_tr(const float* __restrict__ in, float* __restrict__ outp) {
    int i = blockIdx.x * 256 + threadIdx.x;          // < B*C*HW = 1048576
    int b  = i >> 18;                                // C*HW = 262144
    int c  = (i >> 12) & 63;
    int hw = i & 4095;
    outp[(((b << 12) + hw) << 6) + c] = in[i];
}

// ------------------------------------------------------- img (3,HW)->(HW,3)
__global__ void k_imgt(const float* __restrict__ img, float* __restrict__ imgT) {
    int i = blockIdx.x * 256 + threadIdx.x;          // < 3*HWP
    if (i < 3 * HWP) {
        int c = i / HWP, hw = i % HWP;
        imgT[hw * 3 + c] = img[i];
    }
}

// ------------------------------------------- per-pixel norms + self-sim sum
__global__ __launch_bounds__(256) void k_norms(const float* __restrict__ z1t,
                                               const float* __restrict__ z2t,
                                               float* __restrict__ n1sq,
                                               float* __restrict__ n2,
                                               float* __restrict__ s0sum) {
    __shared__ float red[256];
    int g = blockIdx.x * 256 + threadIdx.x;          // < B*HWP
    int b = g >> 12;
    const float4* r1 = reinterpret_cast<const float4*>(z1t) + (size_t)g * 16;
    const float4* r2 = reinterpret_cast<const float4*>(z2t) + (size_t)g * 16;
    float s1 = 0.f, s2 = 0.f;
#pragma unroll
    for (int i = 0; i < 16; ++i) {
        float4 a = r1[i]; s1 += a.x*a.x + a.y*a.y + a.z*a.z + a.w*a.w;
        float4 c = r2[i]; s2 += c.x*c.x + c.y*c.y + c.z*c.z + c.w*c.w;
    }
    n1sq[g] = s1;
    n2[g]   = sqrtf(s2);
    float ss = fminf(fabsf(s1 / fmaxf(s1, EPSV)), 1.0f);  // cos_self clipped
    red[threadIdx.x] = ss;
    __syncthreads();
    for (int s = 128; s > 0; s >>= 1) {
        if (threadIdx.x < s) red[threadIdx.x] += red[threadIdx.x + s];
        __syncthreads();
    }
    if (threadIdx.x == 0) atomicAdd(&s0sum[b], red[0]);
}

// --------------------------------------------------------------- main kernel
// 1 workgroup = (batch b, tile of 16 pixels). Gram rows G[16 x 4096] built in
// 4 chunks of 1024 columns (64KB LDS) via chained v_wmma_f32_16x16x4_f32,
// then consumed by the 256 threads (thread = negative slot k).
__global__ __launch_bounds__(256) void k_main(const float* __restrict__ z1t,
                                              const float* __restrict__ z2t,
                                              const float* __restrict__ imgT,
                                              const float* __restrict__ n1sq,
                                              const float* __restrict__ n2,
                                              const int*   __restrict__ idxh,
                                              const int*   __restrict__ idxw,
                                              float*       __restrict__ snegsum) {
    __shared__ float G[16 * 1024];                   // 64 KB chunk of Gram rows
    const int tile = blockIdx.x & 255;               // HWP/16 tiles
    const int b    = blockIdx.x >> 8;
    const int lane = threadIdx.x & 31;
    const int wave = threadIdx.x >> 5;               // 0..7
    const int m     = lane & 15;
    const int khalf = (lane >> 4) << 1;              // 0 or 2 (A/B f32 K split)

    // A fragments: 16 pixels x 64 channels, kept in registers, reused by all
    // 32 N-tiles this wave produces.  A[m][k], k = 4*kk + khalf + {0,1}.
    const float* z1row = z1t + ((size_t)(b * HWP + tile * 16 + m)) * CC + khalf;
    v2f afr[16];
#pragma unroll
    for (int kk = 0; kk < 16; ++kk)
        afr[kk] = *reinterpret_cast<const v2f*>(z1row + kk * 4);

    // This thread's 16 negative locations (pixel i of tile, slot k=tid).
    const int k = threadIdx.x;
    int locs[16];
#pragma unroll
    for (int i = 0; i < 16; ++i) {
        size_t off = ((size_t)(b * HWP + tile * 16 + i)) * NEGK + k;
        locs[i] = (idxh[off] << 6) | idxw[off];
    }

    const float* z2b = z2t + (size_t)b * HWP * CC;
    float kacc = 0.0f;

    for (int ch = 0; ch < 4; ++ch) {
        // ---- Phase B: dense WMMA, wave w makes N-tiles [w*8, w*8+8) of chunk
#pragma unroll 1
        for (int t = 0; t < 8; ++t) {
            int nt = (ch << 6) + (wave << 3) + t;    // global 16-col tile id
            const float* bcol = z2b + (size_t)(nt * 16 + m) * CC + khalf;
            v2f bfr[16];
#pragma unroll
            for (int kk = 0; kk < 16; ++kk)          // B[k][n] = z2t[n][k]
                bfr[kk] = *reinterpret_cast<const v2f*>(bcol + kk * 4);
            v8f acc = {};
#pragma unroll
            for (int kk = 0; kk < 16; ++kk)
                acc = __builtin_amdgcn_wmma_f32_16x16x4_f32(
                    false, afr[kk], false, bfr[kk], (short)0, acc, false, false);
            // D[m][n]: vgpr r -> M = r + (lane>=16)*8, N = lane&15
            int nloc = ((nt * 16) & 1023) + (lane & 15);
            int mo   = (lane >> 4) << 3;
#pragma unroll
            for (int r = 0; r < 8; ++r)
                G[(mo + r) * 1024 + nloc] = acc[r];
        }
        __syncthreads();

        // ---- Phase C: consume negatives whose column is in this chunk
        int lo = ch << 10;
        for (int i = 0; i < 16; ++i) {
            int loc = locs[i];
            int rel = loc - lo;
            if ((unsigned)rel < 1024u) {
                int   p    = tile * 16 + i;
                float dot  = G[i * 1024 + rel];
                float nn1  = sqrtf(n1sq[b * HWP + p]);
                float nn2  = n2[b * HWP + loc];
                float cosv = dot / fmaxf(nn1 * nn2, EPSV);
                float ph = (float)(p >> 6),   pw = (float)(p & 63);
                float ih = (float)(loc >> 6), iw = (float)(loc & 63);
                float dh = ph - ih, dw = pw - iw;
                float euc = sqrtf(dh * dh + dw * dw) * INV_MAX_EUC;
                float dr = imgT[p * 3 + 0] - imgT[loc * 3 + 0];
                float dg = imgT[p * 3 + 1] - imgT[loc * 3 + 1];
                float db = imgT[p * 3 + 2] - imgT[loc * 3 + 2];
                float rgb = sqrtf(dr * dr + dg * dg + db * db) * INV_MAX_RGB;
                float wgt = euc * 0.8f + rgb * 0.2f;
                kacc += fminf(fabsf(cosv * wgt), 1.0f);
            }
        }
        __syncthreads();
    }
    atomicAdd(&snegsum[b * NEGK + k], kacc);
}

// ----------------------------------------------------------------- finalize
__global__ __launch_bounds__(256) void k_final(const float* __restrict__ s0sum,
                                               const float* __restrict__ snegsum,
                                               float* __restrict__ out) {
    __shared__ float red[256];
    __shared__ float bsum[BB];
    int k = threadIdx.x;
    float tot = 0.0f;
    for (int b = 0; b < BB; ++b) {
        float sneg = snegsum[b * NEGK + k] * (1.0f / HWP) * (1.0f / TEMPR);
        tot += sneg;
        float lg = fmaxf(logf(1.0f - sneg), -100.0f);
        red[k] = lg;
        __syncthreads();
        for (int s = 128; s > 0; s >>= 1) {
            if (k < s) red[k] += red[k + s];
            __syncthreads();
        }
        if (k == 0) bsum[b] = red[0];
        __syncthreads();
    }
    red[k] = tot;
    __syncthreads();
    for (int s = 128; s > 0; s >>= 1) {
        if (k < s) red[k] += red[k + s];
        __syncthreads();
    }
    if (k == 0) {
        float totall = red[0];
        float loss = 0.0f, s0a = 0.0f;
        for (int b = 0; b < BB; ++b) {
            float s0  = s0sum[b] / (float)HWP;
            float ls0 = fmaxf(logf(s0), -100.0f);
            loss += -(ls0 + bsum[b]) / (float)(NEGK + 1);
            s0a  += s0;
        }
        out[0] = loss * (1.0f / BB);
        out[1] = s0a * (1.0f / BB);
        out[2] = totall * TEMPR / (float)NEGK * (1.0f / BB);
    }
}

extern "C" void kernel_launch(void* const* d_in, const int* in_sizes, int n_in,
                              void* d_out, int out_size, void* d_ws, size_t ws_size,
                              hipStream_t stream) {
    const float* v1  = (const float*)d_in[0];   // views_1 (4,64,64,64)
    const float* v2  = (const float*)d_in[1];   // views_2 (4,64,64,64)
    const float* img = (const float*)d_in[2];   // img (1,3,64,64)
    const int*   ih  = (const int*)d_in[3];     // (4,4096,256)
    const int*   iw  = (const int*)d_in[4];     // (4,4096,256)
    float* out = (float*)d_out;                 // 3 scalars

    // workspace layout (floats): ~8.6 MB total
    float* ws      = (float*)d_ws;
    float* z1t     = ws;                        // 1048576
    float* z2t     = z1t + (size_t)BB * HWP * CC;
    float* imgT    = z2t + (size_t)BB * HWP * CC;   // 12288
    float* n1sq    = imgT + 3 * HWP;            // 16384
    float* n2v     = n1sq + BB * HWP;           // 16384
    float* s0sum   = n2v + BB * HWP;            // 8 (4 used)
    float* snegsum = s0sum + 8;                 // 1024

    k_zero <<<5, 256, 0, stream>>>(s0sum, 8 + BB * NEGK);
    k_tr   <<<4096, 256, 0, stream>>>(v1, z1t);
    k_tr   <<<4096, 256, 0, stream>>>(v2, z2t);
    k_imgt <<<48, 256, 0, stream>>>(img, imgT);
    k_norms<<<(BB * HWP) / 256, 256, 0, stream>>>(z1t, z2t, n1sq, n2v, s0sum);
    k_main <<<BB * (HWP / 16), 256, 0, stream>>>(z1t, z2t, imgT, n1sq, n2v, ih, iw, snegsum);
    k_final<<<1, 256, 0, stream>>>(s0sum, snegsum, out);
    (void)in_sizes; (void)n_in; (void)out_size; (void)ws_size;
}